// HGCNN_35476429864974
// MI455X (gfx1250) — compile-verified
//
#include <hip/hip_runtime.h>
#include <stdint.h>

typedef float v2f __attribute__((ext_vector_type(2)));
typedef float v8f __attribute__((ext_vector_type(8)));

#define MINN 1e-15f
#define MAXN 0.996f              /* (1 - 4e-3)/sqrt(c), c = 1 */
#define ATLIM (1.0f - 1e-7f)

__device__ __forceinline__ float wsum(float v) {
#pragma unroll
  for (int m = 16; m > 0; m >>= 1) v += __shfl_xor(v, m, 32);
  return v;
}
__device__ __forceinline__ float clamp_at(float x) {
  return fminf(fmaxf(x, -ATLIM), ATLIM);
}

/* ---------------- WMMA fp32 GEMM: mx[N,64] = x[N,128] @ W[64,128]^T -------- */
__global__ void k_gemm(const float* __restrict__ x, const float* __restrict__ W,
                       float* __restrict__ mx, int N) {
  const int wave = threadIdx.x >> 5;
  const int lane = threadIdx.x & 31;
  const int rowTile = blockIdx.x * 2 + (wave >> 2);  // 2 row tiles / block
  const int colTile = wave & 3;                      // 4 col tiles (64 cols)
  const int row0 = rowTile * 16;
  if (row0 >= N) return;                             // wave-uniform exit
  const int m = lane & 15;
  const int khalf = (lane >> 4) * 2;                 // 0 or 2
  int row = row0 + m; if (row >= N) row = N - 1;     // clamp, keep EXEC full
  const int col = colTile * 16 + m;                  // B lane's N index
  const float* xr = x + (size_t)row * 128;
  const float* wr = W + (size_t)col * 128;

  v8f acc = {0.f, 0.f, 0.f, 0.f, 0.f, 0.f, 0.f, 0.f};
#pragma unroll
  for (int k0 = 0; k0 < 128; k0 += 4) {
    v2f a, b;
    a.x = xr[k0 + khalf];     a.y = xr[k0 + khalf + 1];
    b.x = wr[k0 + khalf];     b.y = wr[k0 + khalf + 1];
    acc = __builtin_amdgcn_wmma_f32_16x16x4_f32(false, a, false, b,
                                                (short)0, acc, false, false);
  }
  const int mbase = (lane >> 4) * 8;
#pragma unroll
  for (int r = 0; r < 8; ++r) {
    const int rr = row0 + mbase + r;
    if (rr < N) mx[(size_t)rr * 64 + colTile * 16 + m] = acc[r];
  }
}

/* ---------------- constants: hb = proj(expmap0(b)), hbb scalar ------------- */
__global__ void k_const(const float* __restrict__ b, const float* __restrict__ bb,
                        float* __restrict__ hb, float* __restrict__ hbb) {
  __shared__ float red[64];
  const int t = threadIdx.x;
  const float v = b[t];
  red[t] = v * v; __syncthreads();
  for (int o = 32; o > 0; o >>= 1) { if (t < o) red[t] += red[t + o]; __syncthreads(); }
  const float s2 = red[0];
  const float sq = sqrtf(s2);
  const float un = fmaxf(sq, MINN);
  const float tt = tanhf(un) / un;
  const float hn = fmaxf(tt * sq, MINN);
  const float f = (hn > MAXN) ? MAXN / hn : 1.f;
  hb[t] = f * tt * v;
  if (t == 0) {
    const float u = bb[0];
    const float au = fmaxf(fabsf(u), MINN);
    float hh = tanhf(au) * u / au;
    const float n = fmaxf(fabsf(hh), MINN);
    if (n > MAXN) hh *= MAXN / n;
    hbb[0] = hh;
  }
}

/* ------- per-node (wave) hyp_linear tail + logmap0 -> utx[N,64] ------------ */
__global__ void k_rowA(const float* __restrict__ x, const float* __restrict__ mx,
                       const float* __restrict__ hb, float* __restrict__ utx, int N) {
  const int wave = threadIdx.x >> 5, lane = threadIdx.x & 31;
  const int i = blockIdx.x * 8 + wave;
  if (i >= N) return;
  float s = 0.f;
#pragma unroll
  for (int q = 0; q < 4; ++q) { float t = x[(size_t)i * 128 + lane + 32 * q]; s += t * t; }
  const float xn = fmaxf(sqrtf(wsum(s)), MINN);
  const float m0 = mx[(size_t)i * 64 + lane];
  const float m1 = mx[(size_t)i * 64 + lane + 32];
  const float mxn2 = wsum(m0 * m0 + m1 * m1);
  const float mxn = fmaxf(sqrtf(mxn2), MINN);
  float scl = tanhf(mxn / xn * atanhf(clamp_at(xn))) / mxn;
  if (mxn2 == 0.f) scl = 0.f;                         /* all(mx==0) guard */
  float r0 = scl * m0, r1 = scl * m1;
  float n = fmaxf(sqrtf(wsum(r0 * r0 + r1 * r1)), MINN);
  if (n > MAXN) { const float f = MAXN / n; r0 *= f; r1 *= f; }
  const float h0 = hb[lane], h1 = hb[lane + 32];
  const float x2 = wsum(r0 * r0 + r1 * r1);
  const float y2 = wsum(h0 * h0 + h1 * h1);
  const float xy = wsum(r0 * h0 + r1 * h1);
  const float ca = 1.f + 2.f * xy + y2, cb = 1.f - x2;
  const float den = fmaxf(1.f + 2.f * xy + x2 * y2, MINN);
  float o0 = (ca * r0 + cb * h0) / den, o1 = (ca * r1 + cb * h1) / den;
  float nn = fmaxf(sqrtf(wsum(o0 * o0 + o1 * o1)), MINN);
  if (nn > MAXN) { const float f = MAXN / nn; o0 *= f; o1 *= f; }
  const float pn = fmaxf(sqrtf(wsum(o0 * o0 + o1 * o1)), MINN);
  const float tl = atanhf(clamp_at(pn)) / pn;
  utx[(size_t)i * 64 + lane] = tl * o0;
  utx[(size_t)i * 64 + lane + 32] = tl * o1;
}

/* ---------------- graph degree / dinv -------------------------------------- */
__global__ void k_deg(const long long* __restrict__ ei, long long E, float* __restrict__ deg) {
  const long long e = blockIdx.x * (long long)blockDim.x + threadIdx.x;
  if (e >= E) return;
  const long long s = ei[e], d = ei[E + e];
  if (s != d) atomicAdd(&deg[s], 1.0f);
}
__global__ void k_dinv(const float* __restrict__ deg, float* __restrict__ dinv, int N) {
  const int i = blockIdx.x * blockDim.x + threadIdx.x;
  if (i >= N) return;
  const float dg = deg[i];
  dinv[i] = (dg > 0.f) ? 1.f / sqrtf(dg) : 0.f;
}

/* ---------------- neigh[dst] += dinv[s]*dinv[d]*utx[src] ------------------- */
__global__ void k_neigh(const long long* __restrict__ ei, long long E,
                        const float* __restrict__ dinv, const float* __restrict__ utx,
                        float* __restrict__ nb) {
  const long long gid = blockIdx.x * (long long)blockDim.x + threadIdx.x;
  const long long e = gid >> 1; const int half = (int)(gid & 1);
  if (e >= E) return;
  const long long s = ei[e], d = ei[E + e];
  if (s == d) return;
  const float ne = dinv[s] * dinv[d];
  if (ne == 0.f) return;
  const float* us = utx + (size_t)s * 64 + half * 32;
  float* nd = nb + (size_t)d * 64 + half * 32;
  for (int j = 0; j < 32; ++j) atomicAdd(&nd[j], ne * us[j]);
}

/* ---------------- score = sum |utx - neigh| -------------------------------- */
__global__ void k_score(const float* __restrict__ utx, const float* __restrict__ nb,
                        float* __restrict__ score, int N) {
  const int i = blockIdx.x * blockDim.x + threadIdx.x;
  if (i >= N) return;
  float s = 0.f;
  for (int j = 0; j < 64; ++j) s += fabsf(utx[(size_t)i * 64 + j] - nb[(size_t)i * 64 + j]);
  score[i] = s;
}

/* ---------------- exact top-k threshold via 16+16 radix select ------------- */
__global__ void k_hist1(const float* __restrict__ score, unsigned* __restrict__ h, int N) {
  const int i = blockIdx.x * blockDim.x + threadIdx.x;
  if (i >= N) return;
  const unsigned bits = __float_as_uint(score[i]);   /* scores >= 0 -> monotonic */
  atomicAdd(&h[bits >> 16], 1u);
}
__global__ void k_findbin(const unsigned* __restrict__ h, int N, int* __restrict__ params) {
  const long long k = (long long)((double)N * 0.75);
  long long cum = 0;
  for (int i = 65535; i >= 0; --i) {
    const unsigned c = h[i];
    if (cum + c >= k) { params[0] = i; params[1] = (int)(k - cum); return; }
    cum += c;
  }
  params[0] = 0; params[1] = 1;
}
__global__ void k_hist2(const float* __restrict__ score, const int* __restrict__ params,
                        unsigned* __restrict__ h, int N) {
  const int i = blockIdx.x * blockDim.x + threadIdx.x;
  if (i >= N) return;
  const unsigned bits = __float_as_uint(score[i]);
  if ((int)(bits >> 16) == params[0]) atomicAdd(&h[bits & 0xffffu], 1u);
}
__global__ void k_findlow(const unsigned* __restrict__ h, const int* __restrict__ params,
                          float* __restrict__ Tval) {
  const int r = params[1];
  long long cum = 0;
  for (int l = 65535; l >= 0; --l) {
    const unsigned c = h[l];
    if (cum + c >= (long long)r) {
      *Tval = __uint_as_float(((unsigned)params[0] << 16) | (unsigned)l);
      return;
    }
    cum += c;
  }
  *Tval = 0.f;
}
__global__ void k_sel(const float* __restrict__ score, const float* __restrict__ Tval,
                      float* __restrict__ sel, int N) {
  const int i = blockIdx.x * blockDim.x + threadIdx.x;
  if (i >= N) return;
  sel[i] = (score[i] > *Tval) ? 1.f : 0.f;
}

/* ------ concat[dst] cols[0:64)=sum(sel*utx), cols[64:128)=sum(utx) --------- */
__global__ void k_sumagg(const long long* __restrict__ ei, long long E,
                         const float* __restrict__ utx, const float* __restrict__ sel,
                         float* __restrict__ cc) {
  const long long gid = blockIdx.x * (long long)blockDim.x + threadIdx.x;
  const long long e = gid >> 1; const int half = (int)(gid & 1);
  if (e >= E) return;
  const long long s = ei[e], d = ei[E + e];
  const float sl = sel[s];
  const float* us = utx + (size_t)s * 64 + half * 32;
  float* cd = cc + (size_t)d * 128 + half * 32;
  for (int j = 0; j < 32; ++j) {
    const float v = us[j];
    atomicAdd(&cd[64 + j], v);
    if (sl != 0.f) atomicAdd(&cd[j], v);
  }
}

/* ------ per-node (wave): g = sigmoid(logmap0(hyp_linear(expmap0(cc)))) * sel */
__global__ void k_gate(const float* __restrict__ cc, const float* __restrict__ betaW,
                       const float* __restrict__ hbb, const float* __restrict__ sel,
                       float* __restrict__ g, int N) {
  const int wave = threadIdx.x >> 5, lane = threadIdx.x & 31;
  const int i = blockIdx.x * 8 + wave;
  if (i >= N) return;
  const float* c = cc + (size_t)i * 128;
  float v[4]; float s2 = 0.f;
#pragma unroll
  for (int q = 0; q < 4; ++q) { v[q] = c[lane + 32 * q]; s2 += v[q] * v[q]; }
  s2 = wsum(s2);
  const float sq = sqrtf(s2);
  const float un = fmaxf(sq, MINN);
  const float t = tanhf(un) / un;
  const float hn_raw = t * sq;
  const float nn = fmaxf(hn_raw, MINN);
  const float f = (nn > MAXN) ? MAXN / nn : 1.f;
  const float tf = t * f;
#pragma unroll
  for (int q = 0; q < 4; ++q) v[q] *= tf;            /* hyp_sums row */
  const float xn = fmaxf(f * hn_raw, MINN);
  float dl = 0.f;
#pragma unroll
  for (int q = 0; q < 4; ++q) dl += v[q] * betaW[lane + 32 * q];
  const float mv = wsum(dl);
  const float mxn = fmaxf(fabsf(mv), MINN);
  float r = tanhf(mxn / xn * atanhf(clamp_at(xn))) * mv / mxn;
  if (mv == 0.f) r = 0.f;
  const float rn = fmaxf(fabsf(r), MINN);
  if (rn > MAXN) r *= MAXN / rn;
  const float y = hbb[0];
  const float x2 = r * r, y2 = y * y, xy = r * y;
  const float num = (1.f + 2.f * xy + y2) * r + (1.f - x2) * y;
  const float den = fmaxf(1.f + 2.f * xy + x2 * y2, MINN);
  float bo = num / den;
  const float bn = fmaxf(fabsf(bo), MINN);
  if (bn > MAXN) bo *= MAXN / bn;
  const float pn = fmaxf(fabsf(bo), MINN);
  const float lb = atanhf(clamp_at(pn)) * bo / pn;
  const float wsel = 1.f / (1.f + expf(-lb));
  if (lane == 0) g[i] = wsel * sel[i];
}

/* ---------------- Ax[dst] += g[src] * utx[src] ----------------------------- */
__global__ void k_ax(const long long* __restrict__ ei, long long E,
                     const float* __restrict__ utx, const float* __restrict__ g,
                     float* __restrict__ ax) {
  const long long gid = blockIdx.x * (long long)blockDim.x + threadIdx.x;
  const long long e = gid >> 1; const int half = (int)(gid & 1);
  if (e >= E) return;
  const long long s = ei[e], d = ei[E + e];
  const float gs = g[s];
  if (gs == 0.f) return;
  const float* us = utx + (size_t)s * 64 + half * 32;
  float* ad = ax + (size_t)d * 64 + half * 32;
  for (int j = 0; j < 32; ++j) atomicAdd(&ad[j], gs * us[j]);
}

/* ---------------- out = proj(expmap0(utx + relu(Ax))) ---------------------- */
__global__ void k_out(const float* __restrict__ utx, const float* __restrict__ ax,
                      float* __restrict__ out, int N) {
  const int i = blockIdx.x * blockDim.x + threadIdx.x;
  if (i >= N) return;
  const float* u = utx + (size_t)i * 64;
  const float* a = ax + (size_t)i * 64;
  float s2 = 0.f;
  for (int j = 0; j < 64; ++j) { const float v = u[j] + fmaxf(a[j], 0.f); s2 += v * v; }
  const float sq = sqrtf(s2);
  const float un = fmaxf(sq, MINN);
  const float t = tanhf(un) / un;
  const float nn = fmaxf(t * sq, MINN);
  const float f = (nn > MAXN) ? MAXN / nn : 1.f;
  const float tf = t * f;
  for (int j = 0; j < 64; ++j) out[(size_t)i * 64 + j] = tf * (u[j] + fmaxf(a[j], 0.f));
}

extern "C" void kernel_launch(void* const* d_in, const int* in_sizes, int n_in,
                              void* d_out, int out_size, void* d_ws, size_t ws_size,
                              hipStream_t stream) {
  const float* x       = (const float*)d_in[0];
  const float* W       = (const float*)d_in[1];
  const float* b       = (const float*)d_in[2];
  const float* betaW   = (const float*)d_in[3];
  const float* betab   = (const float*)d_in[4];
  const long long* ei  = (const long long*)d_in[5];
  const int N = in_sizes[0] / 128;
  const long long E = (long long)in_sizes[5] / 2;
  float* out = (float*)d_out;

  char* wsb = (char*)d_ws;
  const size_t fN64 = (size_t)N * 64 * sizeof(float);
  float* utx = (float*)(wsb + 0);                 /* persists                   */
  float* nb  = (float*)(wsb + fN64);              /* neigh, then reused as Ax   */
  float* mx  = (float*)(wsb + 2 * fN64);          /* dead after k_rowA          */
  float* cc  = (float*)(wsb + 2 * fN64);          /* overlaps mx; 2*fN64        */
  char* p = wsb + 4 * fN64;
  float* deg   = (float*)p; p += (size_t)N * 4;
  float* dinv  = (float*)p; p += (size_t)N * 4;
  float* score = (float*)p; p += (size_t)N * 4;
  float* sel   = (float*)p; p += (size_t)N * 4;
  float* g     = (float*)p; p += (size_t)N * 4;
  p = (char*)(((uintptr_t)p + 255) & ~(uintptr_t)255);
  float* hb   = (float*)p;                        /* 64 floats */
  float* hbb  = (float*)(p + 256);
  float* Tval = (float*)(p + 260);
  int* params = (int*)(p + 264);
  unsigned* hist1 = (unsigned*)(p + 512);
  unsigned* hist2 = hist1 + 65536;

  const int TB = 256;
  const unsigned gN   = (unsigned)((N + TB - 1) / TB);
  const unsigned gE   = (unsigned)((E + TB - 1) / TB);
  const unsigned g2E  = (unsigned)((2 * E + TB - 1) / TB);
  const unsigned gW   = (unsigned)((N + 7) / 8);     /* 8 waves per block */
  const unsigned gT   = (unsigned)((N + 31) / 32);   /* 2 row-tiles per block */

  hipMemsetAsync(deg, 0, (size_t)N * 4, stream);
  hipMemsetAsync(nb, 0, fN64, stream);
  hipMemsetAsync(hist1, 0, 2 * 65536 * sizeof(unsigned), stream);

  k_const<<<1, 64, 0, stream>>>(b, betab, hb, hbb);
  k_gemm <<<gT, TB, 0, stream>>>(x, W, mx, N);
  k_rowA <<<gW, TB, 0, stream>>>(x, mx, hb, utx, N);
  k_deg  <<<gE, TB, 0, stream>>>(ei, E, deg);
  k_dinv <<<gN, TB, 0, stream>>>(deg, dinv, N);
  k_neigh<<<g2E, TB, 0, stream>>>(ei, E, dinv, utx, nb);
  k_score<<<gN, TB, 0, stream>>>(utx, nb, score, N);
  k_hist1<<<gN, TB, 0, stream>>>(score, hist1, N);
  k_findbin<<<1, 1, 0, stream>>>(hist1, N, params);
  k_hist2<<<gN, TB, 0, stream>>>(score, params, hist2, N);
  k_findlow<<<1, 1, 0, stream>>>(hist2, params, Tval);
  k_sel  <<<gN, TB, 0, stream>>>(score, Tval, sel, N);
  hipMemsetAsync(cc, 0, 2 * fN64, stream);        /* mx is dead now */
  k_sumagg<<<g2E, TB, 0, stream>>>(ei, E, utx, sel, cc);
  k_gate <<<gW, TB, 0, stream>>>(cc, betaW, hbb, sel, g, N);
  hipMemsetAsync(nb, 0, fN64, stream);            /* reuse as Ax */
  k_ax   <<<g2E, TB, 0, stream>>>(ei, E, utx, g, nb);
  k_out  <<<gN, TB, 0, stream>>>(utx, nb, out, N);
}